// TokenEmbedding_14654428414483
// MI455X (gfx1250) — compile-verified
//
#include <hip/hip_runtime.h>
#include <hip/hip_bf16.h>

#define DIM   256
#define HALF  128
#define B_    32
#define S_    4096
#define NTOK  (B_ * S_)
#define UMAX  4096
#define MAXD  13

typedef float v2f __attribute__((ext_vector_type(2)));
typedef float v4f __attribute__((ext_vector_type(4)));
typedef float v8f __attribute__((ext_vector_type(8)));

// ---------------- init: flags = 0, uniq = S (fill value) ----------------
__global__ void k_init(int* __restrict__ flags, int* __restrict__ uniq) {
    int i = blockIdx.x * blockDim.x + threadIdx.x;
    if (i < UMAX) { flags[i] = 0; uniq[i] = S_; }
}

// ---------------- mark present positions ----------------
__global__ void k_flag(const int* __restrict__ npos, int* __restrict__ flags) {
    int i = blockIdx.x * blockDim.x + threadIdx.x;
    if (i < NTOK) flags[npos[i]] = 1;   // races benign (all write 1)
}

// ---------------- exclusive prefix over 4096 flags -> rank; scatter uniq ----------------
// rank[v] == searchsorted(uniq, v) == index of v in uniq when present.
__global__ void k_scan(const int* __restrict__ flags, int* __restrict__ rank,
                       int* __restrict__ uniq) {
    __shared__ int sums[1024];
    int tid = threadIdx.x;
    int base = tid * 4;
    int f0 = flags[base + 0], f1 = flags[base + 1];
    int f2 = flags[base + 2], f3 = flags[base + 3];
    sums[tid] = f0 + f1 + f2 + f3;
    __syncthreads();
    for (int off = 1; off < 1024; off <<= 1) {
        int v   = sums[tid];
        int add = (tid >= off) ? sums[tid - off] : 0;
        __syncthreads();
        sums[tid] = v + add;
        __syncthreads();
    }
    int run = (tid == 0) ? 0 : sums[tid - 1];
    int r0 = run, r1 = r0 + f0, r2 = r1 + f1, r3 = r2 + f2;
    rank[base + 0] = r0; rank[base + 1] = r1;
    rank[base + 2] = r2; rank[base + 3] = r3;
    if (f0) uniq[r0] = base + 0;
    if (f1) uniq[r1] = base + 1;
    if (f2) uniq[r2] = base + 2;
    if (f3) uniq[r3] = base + 3;
}

// ---------------- Cayley: Q = (I - A/2)^{-1} (I + A/2), A = tril(X) - tril(X)^T ----
__global__ void k_cayley(const float* __restrict__ prim, float* __restrict__ Q) {
    __shared__ float M[128][128];          // 64 KB
    int b = blockIdx.x;
    int r = threadIdx.x;                   // 0..127, owns row r
    const float* X  = prim + b * HALF * HALF;
    float*       Qb = Q    + b * HALF * HALF;
    for (int j = 0; j < HALF; ++j) {
        float a = 0.f;
        if (r >= j) a += X[r * HALF + j];  // tril (incl diag)
        if (j >= r) a -= X[j * HALF + r];  // - tril^T  (diag cancels)
        float ha  = 0.5f * a;
        float eye = (r == j) ? 1.f : 0.f;
        M[r][j]          = eye - ha;
        Qb[r * HALF + j] = eye + ha;
    }
    __syncthreads();
    for (int i = 0; i < HALF; ++i) {
        float piv = M[i][i];
        __syncthreads();                   // everyone read pivot before row i changes
        float pinv = 1.f / piv;
        M[i][r]          *= pinv;          // thread r scales column r of row i
        Qb[i * HALF + r] *= pinv;
        __syncthreads();
        if (r != i) {
            float f = M[r][i];
            for (int j = 0; j < HALF; ++j) {
                M[r][j]          -= f * M[i][j];
                Qb[r * HALF + j] -= f * Qb[i * HALF + j];
            }
        }
        __syncthreads();
    }
}

// ---------------- path embedding scan ----------------
// Block = 128 threads (4 waves) owns 16 unique rows; wave w computes N-tiles {2w, 2w+1}.
// Q0/Q1 staged once into LDS via async-to-LDS copies; V double-buffered in LDS.
// V <- V @ Q[bit]^T per depth (conditionally per row) via v_wmma_f32_16x16x4_f32.
__global__ void __launch_bounds__(128)
k_pathemb(const int* __restrict__ uniq, const float* __restrict__ Q,
          const float* __restrict__ identity, float* __restrict__ pathemb) {
    __shared__ float Qs[2][HALF][HALF];    // 128 KB staged Q0,Q1
    __shared__ float V[2][16][HALF];       // 16 KB double-buffered maps tile
    __shared__ int   poss[16];
    int tid  = threadIdx.x;                // 0..127
    int lane = tid & 31;                   // wave32 lane
    int wv   = tid >> 5;                   // wave id 0..3
    int ub   = blockIdx.x * 16;

    // ---- async-stage Q (32768 floats) into LDS: 16B per lane per op, coalesced ----
    {
        const float* gq = Q;
        float* qflat = &Qs[0][0][0];
        for (int i = tid * 4; i < 2 * HALF * HALF; i += 128 * 4) {
            unsigned ldsa = (unsigned)(unsigned long long)(qflat + i);
            const float* gp = gq + i;
            asm volatile("global_load_async_to_lds_b128 %0, %1, off"
                         :: "v"(ldsa), "v"(gp) : "memory");
        }
        asm volatile("s_wait_asynccnt 0x0" ::: "memory");
    }
    if (tid < 16) poss[tid] = uniq[ub + tid];
    for (int i = tid; i < 16 * HALF; i += 128)
        V[0][i >> 7][i & 127] = identity[i & 127];
    __syncthreads();

    int half = lane >> 4;                  // which K-half this lane supplies
    int l15  = lane & 15;
    int cur = 0;
    for (int d = 0; d < MAXD; ++d) {
        int nxt = cur ^ 1;
        for (int tt = 0; tt < 2; ++tt) {   // this wave's 2 N-tiles
            int t = wv * 2 + tt;
            v8f acc0 = {};
            v8f acc1 = {};
            int n = t * 16 + l15;          // B-matrix column (row of Q)
            for (int c = 0; c < 32; ++c) { // K = 128 in chunks of 4
                int k = c * 4 + 2 * half;
                v2f a  = *(const v2f*)&V[cur][l15][k];   // A: 16x4, row=l15
                v2f b0 = *(const v2f*)&Qs[0][n][k];      // B[k][n] = Q[n][k]
                v2f b1 = *(const v2f*)&Qs[1][n][k];
                acc0 = __builtin_amdgcn_wmma_f32_16x16x4_f32(
                           false, a, false, b0, (short)0, acc0, false, false);
                acc1 = __builtin_amdgcn_wmma_f32_16x16x4_f32(
                           false, a, false, b1, (short)0, acc1, false, false);
            }
            // per-row select by path bit; write into next buffer
            for (int r = 0; r < 8; ++r) {
                int row = r + 8 * half;    // C/D layout: vgpr r, lane-half selects M / M+8
                int col = t * 16 + l15;
                int pos = poss[row];
                bool valid = (pos >> (d + 1)) > 0;
                int  bit   = (pos >> d) & 1;
                float nv = valid ? (bit ? acc1[r] : acc0[r]) : V[cur][row][col];
                V[nxt][row][col] = nv;
            }
        }
        __syncthreads();                   // step boundary across the 4 waves
        cur ^= 1;
    }
    for (int i = tid; i < 16 * HALF; i += 128)
        pathemb[(ub + (i >> 7)) * HALF + (i & 127)] = V[cur][i >> 7][i & 127];
}

// ---------------- final assembly: [B,S,256] = concat(content, pos_enc) ----------------
// One thread per (token, 4 cols): b128 loads/stores, pure HBM bandwidth.
__global__ void k_assemble(const int* __restrict__ tt, const int* __restrict__ tv,
                           const int* __restrict__ np, const float* __restrict__ ew,
                           const int* __restrict__ rank,
                           const float* __restrict__ pathemb, float* __restrict__ out) {
    long long idx = (long long)blockIdx.x * blockDim.x + threadIdx.x;
    if (idx >= (long long)NTOK * (HALF / 4)) return;
    int t = (int)(idx >> 5);               // token
    int c = (int)(idx & 31) * 4;           // column group
    int ty = tt[t];
    int va = tv[t];
    int p  = np[t];
    v4f content = {};
    if (ty == 0)      content = *(const v4f*)(ew + c);
    else if (ty == 1) content = *(const v4f*)(ew + (va + 1) * HALF + c);
    else if (ty == 2) content = *(const v4f*)(ew + (va + 5) * HALF + c);
    else if (ty == 3) { if (va == -1) content = *(const v4f*)(ew + 10 * HALF + c); }
    else if (ty == 4) {
        int v = va < 0 ? 0 : (va > S_ - 1 ? S_ - 1 : va);
        content = *(const v4f*)(pathemb + rank[v] * HALF + c);
    }
    v4f pe = *(const v4f*)(pathemb + rank[p] * HALF + c);
    long long o = (long long)t * DIM + c;
    *(v4f*)(out + o)        = content;
    *(v4f*)(out + o + HALF) = pe;
}

extern "C" void kernel_launch(void* const* d_in, const int* in_sizes, int n_in,
                              void* d_out, int out_size, void* d_ws, size_t ws_size,
                              hipStream_t stream) {
    const int*   token_types  = (const int*)  d_in[0];
    const int*   token_values = (const int*)  d_in[1];
    const int*   node_pos     = (const int*)  d_in[2];
    const float* primitives   = (const float*)d_in[3];
    const float* identity     = (const float*)d_in[4];
    const float* embed_weight = (const float*)d_in[5];
    float* out = (float*)d_out;

    char* ws = (char*)d_ws;
    int*   flags   = (int*)  (ws + 0);                       // 4096 ints
    int*   rank    = (int*)  (ws + 16384);                   // 4096 ints
    int*   uniq    = (int*)  (ws + 32768);                   // 4096 ints
    float* Q       = (float*)(ws + 49152);                   // 2*128*128 floats
    float* pathemb = (float*)(ws + 49152 + 2 * HALF * HALF * sizeof(float)); // 4096*128 f32

    k_init<<<(UMAX + 255) / 256, 256, 0, stream>>>(flags, uniq);
    k_flag<<<(NTOK + 255) / 256, 256, 0, stream>>>(node_pos, flags);
    k_scan<<<1, 1024, 0, stream>>>(flags, rank, uniq);
    k_cayley<<<2, 128, 0, stream>>>(primitives, Q);
    k_pathemb<<<UMAX / 16, 128, 0, stream>>>(uniq, Q, identity, pathemb);
    long long total = (long long)NTOK * (HALF / 4);
    k_assemble<<<(unsigned)((total + 255) / 256), 256, 0, stream>>>(
        token_types, token_values, node_pos, embed_weight, rank, pathemb, out);
}